// GNN_25907242729955
// MI455X (gfx1250) — compile-verified
//
#include <hip/hip_runtime.h>
#include <hip/hip_bf16.h>

#define N_NODES 50000
#define N_EDGES 312500
#define F 256            // IN_F == OUT_F
#define K2 512           // 2*IN_F
#define M_TILES 3125     // N_NODES / 16
#define BSTRIDE 516      // padded LDS row stride (words): 516 % 64 == 4 -> conflict-free

typedef float v2f __attribute__((ext_vector_type(2)));
typedef float v8f __attribute__((ext_vector_type(8)));

// ---------------------------------------------------------------------------
// Zero the workspace (S and deg).
// ---------------------------------------------------------------------------
__global__ void GNN_zero_kernel(float* __restrict__ p, size_t n) {
    size_t i = (size_t)blockIdx.x * blockDim.x + threadIdx.x;
    size_t stride = (size_t)gridDim.x * blockDim.x;
    for (; i < n; i += stride) p[i] = 0.0f;
}

// ---------------------------------------------------------------------------
// Scatter phase: S[Ai[e]] += x[Aj[e]] (row of 256 f32), deg[Ai[e]] += 1.
// One wave (32 lanes) per edge; wave-uniform index loads; coalesced row reads;
// f32 atomics land in L2 (x and S both fit in the 192MB L2).
// ---------------------------------------------------------------------------
__global__ __launch_bounds__(256) void GNN_scatter_kernel(
    const float* __restrict__ x,
    const int*   __restrict__ Ai,
    const int*   __restrict__ Aj,
    float* __restrict__ S,
    float* __restrict__ deg)
{
    const int lane   = threadIdx.x & 31;
    const int wave   = (blockIdx.x * (blockDim.x >> 5)) + (threadIdx.x >> 5);
    const int nwaves = gridDim.x * (blockDim.x >> 5);

    for (int e = wave; e < N_EDGES; e += nwaves) {
        const int ai = Ai[e];
        const int aj = Aj[e];

        // Prefetch next edge's source row (global_prefetch_b8 on gfx1250).
        const int en = e + nwaves;
        if (en < N_EDGES) {
            const int ajn = Aj[en];
            __builtin_prefetch(x + (size_t)ajn * F + lane * 8, 0, 0);
        }

        if (lane == 0) atomicAdd(&deg[ai], 1.0f);

        const float* __restrict__ xr = x + (size_t)aj * F;
        float*       __restrict__ sr = S + (size_t)ai * F;
        #pragma unroll
        for (int c = 0; c < F; c += 32) {
            atomicAdd(&sr[c + lane], xr[c + lane]);
        }
    }
}

// ---------------------------------------------------------------------------
// GEMM phase with native fp32 WMMA (V_WMMA_F32_16X16X4_F32):
//   out[m][n] = sum_k  A[m][k] * W[n][k]  +  deg[m]*b[n]
// where A[m][k] = deg[m]*x[m][k] for k<256, S[m][k-256] for k>=256.
//
// Block = 256 threads = 8 waves. blockIdx.y selects a 16-wide n-tile; the
// W[n0:n0+16, 0:512] slice is staged in LDS once and shared by all 8 waves.
// Each wave owns one 16x16 output tile and runs the K=512 loop in steps of 4
// -> 128 v_wmma_f32_16x16x4_f32 per wave, fp32-exact accumulation.
//
// A-fragment (32-bit A 16x4, ISA layout): lane<16 holds row M=lane with
// {v0:K0, v1:K1}; lane>=16 holds M=lane-16 with {v0:K2, v1:K3}.
// B-fragment (4x16): lane<16 -> N=lane {v0:K0,v1:K1}; lane>=16 -> {v0:K2,v1:K3}.
// D layout: VGPR i: lanes 0-15 -> M=i, N=lane; lanes 16-31 -> M=8+i, N=lane-16.
// ---------------------------------------------------------------------------
__global__ __launch_bounds__(256) void GNN_gemm_kernel(
    const float* __restrict__ x,
    const float* __restrict__ W,
    const float* __restrict__ b,
    const float* __restrict__ S,
    const float* __restrict__ deg,
    float* __restrict__ out)
{
    __shared__ float Bs[16 * BSTRIDE];           // ~33 KB of the 320 KB WGP LDS

    const int n0   = blockIdx.y * 16;
    const int wave = threadIdx.x >> 5;
    const int lane = threadIdx.x & 31;
    const int mtile = blockIdx.x * 8 + wave;

    // Cooperatively stage W[n0+j][k] -> Bs[j*BSTRIDE + k]  (16 x 512 f32)
    for (int i = threadIdx.x; i < 16 * K2; i += 256) {
        const int j = i >> 9;          // n within tile
        const int k = i & (K2 - 1);
        Bs[j * BSTRIDE + k] = W[(size_t)(n0 + j) * K2 + k];
    }
    __syncthreads();

    if (mtile >= M_TILES) return;      // wave-uniform guard (EXEC stays all-1s)

    const int m0    = mtile * 16;
    const int arow  = m0 + (lane & 15);
    const int khalf = (lane >> 4) * 2;           // 0 for lanes 0-15, 2 for 16-31
    const int ncol  = lane & 15;
    const float dsc = deg[arow];

    const float* __restrict__ xr = x + (size_t)arow * F;
    const float* __restrict__ sr = S + (size_t)arow * F;

    v8f acc = {};

    // K = 0..255 : A = deg*x, B = W1 = W[:, 0:256]
    #pragma unroll 4
    for (int k = 0; k < F; k += 4) {
        v2f a, bb;
        a.x  = dsc * xr[k + khalf];
        a.y  = dsc * xr[k + khalf + 1];
        bb.x = Bs[ncol * BSTRIDE + k + khalf];
        bb.y = Bs[ncol * BSTRIDE + k + khalf + 1];
        acc = __builtin_amdgcn_wmma_f32_16x16x4_f32(
            false, a, false, bb, (short)0, acc, false, false);
    }
    // K = 256..511 : A = S, B = W2 = W[:, 256:512]
    #pragma unroll 4
    for (int k = 0; k < F; k += 4) {
        v2f a, bb;
        a.x  = sr[k + khalf];
        a.y  = sr[k + khalf + 1];
        bb.x = Bs[ncol * BSTRIDE + F + k + khalf];
        bb.y = Bs[ncol * BSTRIDE + F + k + khalf + 1];
        acc = __builtin_amdgcn_wmma_f32_16x16x4_f32(
            false, a, false, bb, (short)0, acc, false, false);
    }

    // Epilogue: out[m][n] = acc + deg[m] * b[n]
    const int nout  = n0 + (lane & 15);
    const int mbase = m0 + (lane >> 4) * 8;
    const float bias = b[nout];
    #pragma unroll
    for (int i = 0; i < 8; ++i) {
        const int m = mbase + i;
        out[(size_t)m * F + nout] = acc[i] + deg[m] * bias;
    }
}

// ---------------------------------------------------------------------------
// Launch: inputs in setup_inputs() order: x, W, b, Ai, Aj.
// Workspace: S [N_NODES*256 f32] then deg [N_NODES f32]  (~51.4 MB).
// ---------------------------------------------------------------------------
extern "C" void kernel_launch(void* const* d_in, const int* in_sizes, int n_in,
                              void* d_out, int out_size, void* d_ws, size_t ws_size,
                              hipStream_t stream) {
    const float* x  = (const float*)d_in[0];
    const float* W  = (const float*)d_in[1];
    const float* b  = (const float*)d_in[2];
    const int*   Ai = (const int*)d_in[3];
    const int*   Aj = (const int*)d_in[4];
    float* out = (float*)d_out;

    float* S   = (float*)d_ws;
    float* deg = S + (size_t)N_NODES * F;

    const size_t nzero = (size_t)N_NODES * F + N_NODES;
    GNN_zero_kernel<<<1024, 256, 0, stream>>>(S, nzero);

    GNN_scatter_kernel<<<2048, 256, 0, stream>>>(x, Ai, Aj, S, deg);

    dim3 grid((M_TILES + 7) / 8, F / 16);        // 391 x 16 blocks
    GNN_gemm_kernel<<<grid, 256, 0, stream>>>(x, W, b, S, deg, out);
}